// PrototypeExplanationHead_82729660055723
// MI455X (gfx1250) — compile-verified
//
#include <hip/hip_runtime.h>
#include <hip/hip_bf16.h>

typedef __attribute__((ext_vector_type(2))) float v2f;
typedef __attribute__((ext_vector_type(8))) float v8f;
typedef __attribute__((ext_vector_type(4))) int   i32x4;

#define B_ROWS 8192
#define D_DIM  512
#define P_ROWS 32768
#define TOPK   3

#define BM 128
#define BN 128
#define BK 32
#define NTILES (P_ROWS / BN)      // 256 column tiles
#define LDS_STRIDE 36             // pad 32 -> 36 floats: conflict-free ds_load_b64 frag reads

// gfx1250 async global->LDS path (guarded: falls back to sync staging if the
// toolchain doesn't declare these builtins)
#if defined(__has_builtin)
#if __has_builtin(__builtin_amdgcn_global_load_async_to_lds_b128) && \
    __has_builtin(__builtin_amdgcn_s_wait_asynccnt)
#define USE_ASYNC_LDS 1
#endif
#endif
#ifndef USE_ASYNC_LDS
#define USE_ASYNC_LDS 0
#endif

#if USE_ASYNC_LDS
// builtin signature (from clang diagnostic): arg0 = int4 AS(1)* (global src),
// arg1 = LDS dst, arg2 = imm offset, arg3 = cpol
typedef __attribute__((address_space(1))) i32x4* gptr_b128;
typedef __attribute__((address_space(3))) i32x4* lptr_b128;
#define NBUF 2
#else
#define NBUF 1
#endif

// ---------------------------------------------------------------------------
// Kernel 1: per-row inverse L2 norm  (inv = 1 / max(||x||, 1e-12))
// ---------------------------------------------------------------------------
__global__ void rownorm_inv_kernel(const float* __restrict__ x,
                                   float* __restrict__ inv, int nrows) {
    int row  = blockIdx.x * 8 + (threadIdx.x >> 5);
    int lane = threadIdx.x & 31;
    if (row >= nrows) return;
    const float4* p = (const float4*)(x + (size_t)row * D_DIM);
    float s = 0.f;
#pragma unroll
    for (int i = lane; i < D_DIM / 4; i += 32) {
        float4 v = p[i];
        s += v.x * v.x + v.y * v.y + v.z * v.z + v.w * v.w;
    }
#pragma unroll
    for (int off = 16; off > 0; off >>= 1) s += __shfl_xor(s, off, 32);
    if (lane == 0) inv[row] = 1.0f / fmaxf(sqrtf(s), 1e-12f);
}

// ---------------------------------------------------------------------------
__device__ __forceinline__ void top3_insert(float v, int i,
                                            float& v0, int& i0,
                                            float& v1, int& i1,
                                            float& v2, int& i2) {
    if (v > v2) {
        if (v > v1) {
            if (v > v0) { v2 = v1; i2 = i1; v1 = v0; i1 = i0; v0 = v; i0 = i; }
            else        { v2 = v1; i2 = i1; v1 = v;  i1 = i; }
        } else          { v2 = v;  i2 = i; }
    }
}

// ---------------------------------------------------------------------------
// Kernel 2: sim = (Z Pᵀ)·invZ[m]·invP[n] via V_WMMA_F32_16X16X4_F32.
// 128x128 block tile, BK=32, 8 waves (4Mx2N), wave tile 32x64 = 2x4 frags.
// Optional fused per-block top-3 partials (FUSE_TOPK).
// ---------------------------------------------------------------------------
template <bool FUSE_TOPK>
__global__ __launch_bounds__(256)
void sim_gemm_wmma_kernel(const float* __restrict__ Z,
                          const float* __restrict__ Pr,
                          const float* __restrict__ invZ,
                          const float* __restrict__ invP,
                          float* __restrict__ sim,
                          float* __restrict__ pval,
                          int*   __restrict__ pidx) {
    __shared__ float As[NBUF][BM * LDS_STRIDE];
    __shared__ float Bs[NBUF][BN * LDS_STRIDE];

    const int tid   = threadIdx.x;
    const int wave  = tid >> 5;
    const int lane  = tid & 31;
    const int waveM = (wave >> 1) * 32;   // 0,32,64,96
    const int waveN = (wave & 1) * 64;    // 0,64
    const int lrow  = lane & 15;          // M (A frag) / N (B,C frag)
    const int khalf = (lane >> 4) * 2;    // K pair select

    const int blockM = blockIdx.y * BM;
    const int blockN = blockIdx.x * BN;

    const int rr = tid >> 3;              // 0..31  (row within 32-row batch)
    const int cc = (tid & 7) * 4;         // 0..28  (float4 column)

    v8f acc[2][4] = {};

#if USE_ASYNC_LDS
    // ---- double-buffered async global->LDS pipeline ----------------------
    auto issue_tile = [&](int k0, int buf) {
#pragma unroll
        for (int it = 0; it < 4; ++it) {
            int r = rr + it * 32;
            __builtin_amdgcn_global_load_async_to_lds_b128(
                (gptr_b128)(Z + (size_t)(blockM + r) * D_DIM + k0 + cc),
                (lptr_b128)&As[buf][r * LDS_STRIDE + cc], 0, 0);
            __builtin_amdgcn_global_load_async_to_lds_b128(
                (gptr_b128)(Pr + (size_t)(blockN + r) * D_DIM + k0 + cc),
                (lptr_b128)&Bs[buf][r * LDS_STRIDE + cc], 0, 0);
        }
    };
    issue_tile(0, 0);
#endif

    for (int k0 = 0; k0 < D_DIM; k0 += BK) {
#if USE_ASYNC_LDS
        const int buf = (k0 / BK) & 1;
        if (k0 + BK < D_DIM) {
            issue_tile(k0 + BK, buf ^ 1);          // 8 new ops in flight
            __builtin_amdgcn_s_wait_asynccnt(8);   // retire current tile only
        } else {
            __builtin_amdgcn_s_wait_asynccnt(0);
        }
        __syncthreads();                            // all waves' tile landed
        const float* Asb = As[buf];
        const float* Bsb = Bs[buf];
#else
        // ---- synchronous staging fallback --------------------------------
#pragma unroll
        for (int it = 0; it < 4; ++it) {
            int r = rr + it * 32;
            float4 a = *(const float4*)(Z  + (size_t)(blockM + r) * D_DIM + k0 + cc);
            float4 b = *(const float4*)(Pr + (size_t)(blockN + r) * D_DIM + k0 + cc);
            *(float4*)&As[0][r * LDS_STRIDE + cc] = a;
            *(float4*)&Bs[0][r * LDS_STRIDE + cc] = b;
        }
        if (k0 + BK < D_DIM) {
            __builtin_prefetch(Z  + (size_t)(blockM + rr) * D_DIM + k0 + BK + cc, 0, 1);
            __builtin_prefetch(Pr + (size_t)(blockN + rr) * D_DIM + k0 + BK + cc, 0, 1);
        }
        __syncthreads();
        const float* Asb = As[0];
        const float* Bsb = Bs[0];
#endif

        // ---- 8 x K4 WMMA steps ------------------------------------------
#pragma unroll
        for (int kk = 0; kk < BK; kk += 4) {
            v2f a[2], b[4];
#pragma unroll
            for (int mi = 0; mi < 2; ++mi)
                a[mi] = *(const v2f*)&Asb[(waveM + mi * 16 + lrow) * LDS_STRIDE + kk + khalf];
#pragma unroll
            for (int ni = 0; ni < 4; ++ni)
                b[ni] = *(const v2f*)&Bsb[(waveN + ni * 16 + lrow) * LDS_STRIDE + kk + khalf];
#pragma unroll
            for (int mi = 0; mi < 2; ++mi)
#pragma unroll
                for (int ni = 0; ni < 4; ++ni)
                    acc[mi][ni] = __builtin_amdgcn_wmma_f32_16x16x4_f32(
                        false, a[mi], false, b[ni], (short)0, acc[mi][ni],
                        false, false);
        }
        __syncthreads();   // done reading this buffer -> next iter may refill
    }

    // ---- epilogue: scale, store, optional fused per-row top-3 partials ----
    const int rowHi = (lane >> 4) * 8;   // C layout: VGPR v -> M = v + 8*(lane/16)
    float invp[4];
#pragma unroll
    for (int ni = 0; ni < 4; ++ni)
        invp[ni] = invP[blockN + waveN + ni * 16 + lrow];

#pragma unroll
    for (int mi = 0; mi < 2; ++mi) {
#pragma unroll
        for (int v = 0; v < 8; ++v) {
            const int row = blockM + waveM + mi * 16 + rowHi + v;
            const float invz = invZ[row];
            float tv0 = -3.4e38f, tv1 = -3.4e38f, tv2 = -3.4e38f;
            int   ti0 = 0, ti1 = 0, ti2 = 0;
#pragma unroll
            for (int ni = 0; ni < 4; ++ni) {
                const int col = blockN + waveN + ni * 16 + lrow;
                const float val = acc[mi][ni][v] * invz * invp[ni];
                sim[(size_t)row * P_ROWS + col] = val;
                if (FUSE_TOPK)
                    top3_insert(val, col, tv0, ti0, tv1, ti1, tv2, ti2);
            }
            if (FUSE_TOPK) {
                // merge top-3 lists across the 16 lanes of this half-wave
#pragma unroll
                for (int off = 8; off >= 1; off >>= 1) {
                    float ov0 = __shfl_xor(tv0, off, 32); int oi0 = __shfl_xor(ti0, off, 32);
                    float ov1 = __shfl_xor(tv1, off, 32); int oi1 = __shfl_xor(ti1, off, 32);
                    float ov2 = __shfl_xor(tv2, off, 32); int oi2 = __shfl_xor(ti2, off, 32);
                    top3_insert(ov0, oi0, tv0, ti0, tv1, ti1, tv2, ti2);
                    top3_insert(ov1, oi1, tv0, ti0, tv1, ti1, tv2, ti2);
                    top3_insert(ov2, oi2, tv0, ti0, tv1, ti1, tv2, ti2);
                }
                if (lrow == 0) {
                    size_t pb = ((size_t)row * NTILES + blockIdx.x) * TOPK;
                    pval[pb + 0] = tv0; pval[pb + 1] = tv1; pval[pb + 2] = tv2;
                    pidx[pb + 0] = ti0; pidx[pb + 1] = ti1; pidx[pb + 2] = ti2;
                }
            }
        }
    }
}

// ---------------------------------------------------------------------------
// Kernel 3a (fused path): reduce 256 per-tile top-3 partials per row.
// ---------------------------------------------------------------------------
__global__ __launch_bounds__(256)
void topk_reduce_kernel(const float* __restrict__ pval, const int* __restrict__ pidx,
                        int* __restrict__ idxOut, float* __restrict__ valOut) {
    __shared__ float sval[256][3];
    __shared__ int   sidx[256][3];

    const int row = blockIdx.x;
    const int tid = threadIdx.x;                      // one N-tile per thread
    const size_t base = ((size_t)row * NTILES + tid) * TOPK;

    float v0 = pval[base + 0], v1 = pval[base + 1], v2 = pval[base + 2];
    int   i0 = pidx[base + 0], i1 = pidx[base + 1], i2 = pidx[base + 2];

    sval[tid][0] = v0; sidx[tid][0] = i0;
    sval[tid][1] = v1; sidx[tid][1] = i1;
    sval[tid][2] = v2; sidx[tid][2] = i2;
    __syncthreads();

    for (int stride = 128; stride > 0; stride >>= 1) {
        if (tid < stride) {
#pragma unroll
            for (int j = 0; j < 3; ++j)
                top3_insert(sval[tid + stride][j], sidx[tid + stride][j],
                            v0, i0, v1, i1, v2, i2);
            sval[tid][0] = v0; sidx[tid][0] = i0;
            sval[tid][1] = v1; sidx[tid][1] = i1;
            sval[tid][2] = v2; sidx[tid][2] = i2;
        }
        __syncthreads();
    }

    if (tid == 0) {
        idxOut[row * TOPK + 0] = i0; valOut[row * TOPK + 0] = v0;
        idxOut[row * TOPK + 1] = i1; valOut[row * TOPK + 1] = v1;
        idxOut[row * TOPK + 2] = i2; valOut[row * TOPK + 2] = v2;
    }
}

// ---------------------------------------------------------------------------
// Kernel 3b (fallback path): top-3 per row scanning the full sim matrix.
// ---------------------------------------------------------------------------
__global__ __launch_bounds__(256)
void topk_kernel(const float* __restrict__ sim,
                 int* __restrict__ idxOut, float* __restrict__ valOut) {
    __shared__ float sval[256][3];
    __shared__ int   sidx[256][3];

    const int row = blockIdx.x;
    const int tid = threadIdx.x;
    const float* s = sim + (size_t)row * P_ROWS;

    float v0 = -3.4e38f, v1 = -3.4e38f, v2 = -3.4e38f;
    int   i0 = 0, i1 = 0, i2 = 0;
    for (int c = tid; c < P_ROWS; c += 256)
        top3_insert(s[c], c, v0, i0, v1, i1, v2, i2);

    sval[tid][0] = v0; sidx[tid][0] = i0;
    sval[tid][1] = v1; sidx[tid][1] = i1;
    sval[tid][2] = v2; sidx[tid][2] = i2;
    __syncthreads();

    for (int stride = 128; stride > 0; stride >>= 1) {
        if (tid < stride) {
#pragma unroll
            for (int j = 0; j < 3; ++j)
                top3_insert(sval[tid + stride][j], sidx[tid + stride][j],
                            v0, i0, v1, i1, v2, i2);
            sval[tid][0] = v0; sidx[tid][0] = i0;
            sval[tid][1] = v1; sidx[tid][1] = i1;
            sval[tid][2] = v2; sidx[tid][2] = i2;
        }
        __syncthreads();
    }

    if (tid == 0) {
        idxOut[row * TOPK + 0] = i0; valOut[row * TOPK + 0] = v0;
        idxOut[row * TOPK + 1] = i1; valOut[row * TOPK + 1] = v1;
        idxOut[row * TOPK + 2] = i2; valOut[row * TOPK + 2] = v2;
    }
}

// ---------------------------------------------------------------------------
extern "C" void kernel_launch(void* const* d_in, const int* in_sizes, int n_in,
                              void* d_out, int out_size, void* d_ws, size_t ws_size,
                              hipStream_t stream) {
    const float* z  = (const float*)d_in[0];   // [8192, 512]
    const float* pr = (const float*)d_in[1];   // [32768, 512]
    // d_in[2] = topk scalar (=3), compile-time constant here

    float* invZ = (float*)d_ws;
    float* invP = invZ + B_ROWS;

    const size_t normBytes = (size_t)(B_ROWS + P_ROWS) * sizeof(float);
    const size_t partElems = (size_t)B_ROWS * NTILES * TOPK;
    const size_t fusedBytes = normBytes + partElems * (sizeof(float) + sizeof(int));
    const bool   fuse = (ws_size >= fusedBytes);

    float* pval = (float*)((char*)d_ws + normBytes);
    int*   pidx = (int*)(pval + partElems);

    float* sim     = (float*)d_out;                                   // [B, P]
    int*   topkIdx = (int*)((float*)d_out + (size_t)B_ROWS * P_ROWS); // [B, 3] int32
    float* topkVal = (float*)d_out + (size_t)B_ROWS * P_ROWS + (size_t)B_ROWS * TOPK;

    rownorm_inv_kernel<<<B_ROWS / 8, 256, 0, stream>>>(z,  invZ, B_ROWS);
    rownorm_inv_kernel<<<P_ROWS / 8, 256, 0, stream>>>(pr, invP, P_ROWS);

    dim3 grid(P_ROWS / BN, B_ROWS / BM);  // 256 x 64 workgroups
    if (fuse) {
        sim_gemm_wmma_kernel<true><<<grid, 256, 0, stream>>>(z, pr, invZ, invP,
                                                             sim, pval, pidx);
        topk_reduce_kernel<<<B_ROWS, 256, 0, stream>>>(pval, pidx, topkIdx, topkVal);
    } else {
        sim_gemm_wmma_kernel<false><<<grid, 256, 0, stream>>>(z, pr, invZ, invP,
                                                              sim, nullptr, nullptr);
        topk_kernel<<<B_ROWS, 256, 0, stream>>>(sim, topkIdx, topkVal);
    }
}